// HR2HK_64201171141006
// MI455X (gfx1250) — compile-verified
//
#include <hip/hip_runtime.h>
#include <math.h>

// ---------------------------------------------------------------------------
// Problem constants (match reference)
// ---------------------------------------------------------------------------
#define NORB    9
#define NFEAT   58
#define NK      4
#define N_ATOMS 384
#define N_EDGES 6144
#define ALLN    (N_ATOMS * NORB)   // 3456

// ---------------------------------------------------------------------------
// Compile-time orbital-pair maps: feature f -> (row, col, factor) in 9x9 block
// basis l = {0,1,2}, dims {1,3,5}, offsets {0,1,4}; fac 0.5 for i==j pairs
// ---------------------------------------------------------------------------
struct OrbMaps {
    unsigned char row[NFEAT];
    unsigned char col[NFEAT];
    float         fac[NFEAT];
};

__host__ __device__ constexpr OrbMaps make_maps() {
    OrbMaps m{};
    int dims[3] = {1, 3, 5};
    int off[3]  = {0, 1, 4};
    int t = 0;
    for (int i = 0; i < 3; ++i)
        for (int j = i; j < 3; ++j)
            for (int r = 0; r < dims[i]; ++r)
                for (int c = 0; c < dims[j]; ++c) {
                    m.row[t] = (unsigned char)(off[i] + r);
                    m.col[t] = (unsigned char)(off[j] + c);
                    m.fac[t] = (i == j) ? 0.5f : 1.0f;
                    ++t;
                }
    return m;
}
__device__ constexpr OrbMaps MAPS = make_maps();

// ---------------------------------------------------------------------------
// Async-store zero fill: stream zeros LDS -> global with B128 async stores
// (CDNA5 async engine, ASYNCcnt). Builtin takes (AS1 int4*, AS3 int4*, imm, imm).
// ---------------------------------------------------------------------------
typedef __attribute__((ext_vector_type(4))) int v4i;
typedef __attribute__((address_space(1))) v4i gv4i;   // global int4
typedef __attribute__((address_space(3))) v4i lv4i;   // LDS int4

#if __has_builtin(__builtin_amdgcn_global_store_async_from_lds_b128)
#define ZF_ASYNC_BUILTIN 1
#else
#define ZF_ASYNC_ASM 1
#endif

__device__ inline void wait_async_zero() {
#if __has_builtin(__builtin_amdgcn_s_wait_asynccnt)
    __builtin_amdgcn_s_wait_asynccnt(0);
#else
    asm volatile("s_wait_asynccnt 0" ::: "memory");
#endif
}

__global__ void zero_fill_kernel(float* __restrict__ out, long long n_floats) {
    const long long n4     = n_floats >> 2;
    const long long tid    = (long long)blockIdx.x * blockDim.x + threadIdx.x;
    const long long stride = (long long)gridDim.x * blockDim.x;

    __shared__ __align__(16) float lz[256 * 4];   // 16 B of zeros per thread
    float4 z;
    z.x = 0.f; z.y = 0.f; z.z = 0.f; z.w = 0.f;
    ((float4*)lz)[threadIdx.x] = z;
    __syncthreads();                              // DS writes visible before async reads

    float4* o4 = (float4*)out;

#if defined(ZF_ASYNC_BUILTIN)
    lv4i* lp = (lv4i*)((float4*)lz + threadIdx.x);
    for (long long i = tid; i < n4; i += stride) {
        __builtin_amdgcn_global_store_async_from_lds_b128(
            (gv4i*)(o4 + i), lp, /*offset=*/0, /*cpol=*/0);
    }
    wait_async_zero();
#elif defined(ZF_ASYNC_ASM)
    unsigned lofs = (unsigned)(unsigned long long)
        (__attribute__((address_space(3))) char*)((float4*)lz + threadIdx.x);
    for (long long i = tid; i < n4; i += stride) {
        unsigned long long ga = (unsigned long long)(o4 + i);
        asm volatile("global_store_async_from_lds_b128 %0, %1, off"
                     :: "v"(ga), "v"(lofs) : "memory");
    }
    wait_async_zero();
#else
    for (long long i = tid; i < n4; i += stride) o4[i] = z;
#endif

    // tail (out_size is a multiple of 8 here, but be safe)
    if (tid == 0)
        for (long long t = (n4 << 2); t < n_floats; ++t) out[t] = 0.f;
}

// ---------------------------------------------------------------------------
// Onsite scatter: real symmetric diagonal atom blocks, replicated over k.
// Contribution w at (r,c) and (c,r) == B + conj(B^T) for real values.
// ---------------------------------------------------------------------------
__global__ void onsite_scatter_kernel(const float* __restrict__ ons,
                                      float* __restrict__ out) {
    const int a = blockIdx.x;
    const int f = threadIdx.x;
    if (f >= NFEAT) return;

    const float  w = MAPS.fac[f] * ons[a * NFEAT + f];
    const size_t r = (size_t)a * NORB + MAPS.row[f];
    const size_t c = (size_t)a * NORB + MAPS.col[f];

    for (int k = 0; k < NK; ++k) {
        const size_t b0 = (((size_t)k * ALLN + r) * ALLN + c) * 2;
        const size_t b1 = (((size_t)k * ALLN + c) * ALLN + r) * 2;
        atomicAdd(out + b0, w);   // real part; imag contribution is zero
        atomicAdd(out + b1, w);
    }
}

// ---------------------------------------------------------------------------
// Hopping scatter: v = phase(k,e) * fac * feat at (i*9+R, j*9+C),
// plus conj(v) at the transposed position.  phase = cos(th) - i sin(th),
// th = 2*pi * (k . cell_shift).
// ---------------------------------------------------------------------------
__global__ void hop_scatter_kernel(const float* __restrict__ hop,
                                   const int*   __restrict__ eidx,
                                   const int*   __restrict__ shift,
                                   const float* __restrict__ kpts,
                                   float*       __restrict__ out) {
    const int e = blockIdx.x;
    const int f = threadIdx.x;
    if (f >= NFEAT) return;

    const int ai = eidx[e];
    const int aj = eidx[N_EDGES + e];
    const float sx = (float)shift[e * 3 + 0];
    const float sy = (float)shift[e * 3 + 1];
    const float sz = (float)shift[e * 3 + 2];

    const float  w = MAPS.fac[f] * hop[e * NFEAT + f];
    const size_t r = (size_t)ai * NORB + MAPS.row[f];
    const size_t c = (size_t)aj * NORB + MAPS.col[f];

    const float TWO_PI = 6.28318530717958647692f;

    for (int k = 0; k < NK; ++k) {
        const float dot = kpts[k * 3 + 0] * sx +
                          kpts[k * 3 + 1] * sy +
                          kpts[k * 3 + 2] * sz;
        const float th = TWO_PI * dot;
        const float s  = sinf(th);
        const float cn = cosf(th);
        const float re =  cn * w;
        const float im = -s  * w;

        const size_t b0 = (((size_t)k * ALLN + r) * ALLN + c) * 2;
        const size_t b1 = (((size_t)k * ALLN + c) * ALLN + r) * 2;
        atomicAdd(out + b0,     re);
        atomicAdd(out + b0 + 1, im);
        atomicAdd(out + b1,     re);   // conj for Hermitian partner
        atomicAdd(out + b1 + 1, -im);
    }
}

// ---------------------------------------------------------------------------
// Launch
// ---------------------------------------------------------------------------
extern "C" void kernel_launch(void* const* d_in, const int* in_sizes, int n_in,
                              void* d_out, int out_size, void* d_ws, size_t ws_size,
                              hipStream_t stream) {
    const float* hop   = (const float*)d_in[0];   // [6144, 58]
    const float* ons   = (const float*)d_in[1];   // [384, 58]
    const float* kpts  = (const float*)d_in[2];   // [4, 3]
    const int*   eidx  = (const int*)d_in[3];     // [2, 6144]
    const int*   shift = (const int*)d_in[4];     // [6144, 3]
    float*       out   = (float*)d_out;           // [4, 3456, 3456] complex64 -> 2 floats each

    (void)in_sizes; (void)n_in; (void)d_ws; (void)ws_size;

    // 1) zero 382 MB of output via async LDS->global B128 streaming stores
    zero_fill_kernel<<<8192, 256, 0, stream>>>(out, (long long)out_size);

    // 2) onsite diagonal blocks (k-independent, real symmetric)
    onsite_scatter_kernel<<<N_ATOMS, 64, 0, stream>>>(ons, out);

    // 3) phase-weighted Hermitian hopping scatter
    hop_scatter_kernel<<<N_EDGES, 64, 0, stream>>>(hop, eidx, shift, kpts, out);
}